// MambaTimeSeriesClassifier_30064771072279
// MI455X (gfx1250) — compile-verified
//
#include <hip/hip_runtime.h>
#include <math.h>

typedef float v2f __attribute__((ext_vector_type(2)));
typedef float v8f __attribute__((ext_vector_type(8)));

#define D_MODEL 192
#define NHEADS  24
#define HEADDIM 32
#define D_STATE 128
#define NPOS    256
#define PACKC   928      // 896 xBC channels + 24 dt (+8 zero pad)
#define XB_CH   896
#define BATCH   16
#define TSTEPS  8
#define NCHUNK  (BATCH*TSTEPS)

// workspace layout (float offsets)
#define OFF_PE      0
#define OFF_WP      (OFF_PE + TSTEPS*D_MODEL)
#define OFF_PADPROJ (OFF_WP + PACKC*D_MODEL)
#define OFF_NEGA    (OFF_PADPROJ + TSTEPS*PACKC)
#define OFF_DTV     (OFF_NEGA + 32)
#define OFF_LOGDA   (OFF_DTV + NCHUNK*NPOS*NHEADS)
#define OFF_W       (OFF_LOGDA + NCHUNK*NPOS*NHEADS)
#define OFF_GINT    (OFF_W + NCHUNK*NPOS*NHEADS)
#define OFF_GHEAD   (OFF_GINT + NCHUNK*HEADDIM*D_STATE)

static __device__ __forceinline__ float siluf(float v) {
    return v / (1.f + expf(-v));
}

// ---------------- K0: PE table, packed weights, pad projections, -exp(A_log)
__global__ void k0_setup(const float* __restrict__ in_proj_w,
                         const float* __restrict__ A_log,
                         float* __restrict__ ws) {
    float* pe      = ws + OFF_PE;
    float* Wp      = ws + OFF_WP;
    float* padproj = ws + OFF_PADPROJ;
    float* negA    = ws + OFF_NEGA;
    int tid = threadIdx.x;
    const float kln = logf(10000.0f) / (float)D_MODEL;
    for (int idx = tid; idx < TSTEPS*D_MODEL; idx += blockDim.x) {
        int t = idx / D_MODEL, d = idx % D_MODEL;
        float div = expf(-(float)((d >> 1) << 1) * kln);
        float ang = (float)t * div;
        pe[idx] = (d & 1) ? cosf(ang) : sinf(ang);
    }
    for (int e = tid; e < PACKC*D_MODEL; e += blockDim.x) {
        int cp = e / D_MODEL, k = e % D_MODEL;
        float v = 0.f;
        if (cp < XB_CH)               v = in_proj_w[(768 + cp)*D_MODEL + k];
        else if (cp < XB_CH + NHEADS) v = in_proj_w[(768 + 1024 + (cp - XB_CH))*D_MODEL + k];
        Wp[e] = v;
    }
    if (tid < NHEADS) negA[tid] = -expf(A_log[tid]);
    __syncthreads();
    for (int it = tid; it < TSTEPS*PACKC; it += blockDim.x) {
        int t = it / PACKC, cp = it % PACKC;
        const float* wr = Wp + cp*D_MODEL;
        const float* pr = pe + t*D_MODEL;
        float a0 = 0.f, a1 = 0.f, a2 = 0.f, a3 = 0.f;
        for (int k = 0; k < D_MODEL; k += 4) {
            a0 += wr[k  ] * pr[k  ];
            a1 += wr[k+1] * pr[k+1];
            a2 += wr[k+2] * pr[k+2];
            a3 += wr[k+3] * pr[k+3];
        }
        padproj[it] = (a0 + a1) + (a2 + a3);
    }
}

// ---------------- K1: dt projection -> softplus -> logdA
__global__ void k1_dt(const float* __restrict__ inp,
                      const float* __restrict__ dt_bias,
                      float* __restrict__ ws) {
    const float* pe   = ws + OFF_PE;
    const float* Wp   = ws + OFF_WP;
    const float* negA = ws + OFF_NEGA;
    float* dtv   = ws + OFF_DTV;
    float* logdA = ws + OFF_LOGDA;
    int chunk = blockIdx.x;
    int b = chunk >> 3, tp = chunk & 7;
    for (int item = threadIdx.x; item < NPOS*NHEADS; item += blockDim.x) {
        int s = item / NHEADS, h = item % NHEADS;
        const float* wr = Wp + (XB_CH + h)*D_MODEL;
        const float* pr = pe + tp*D_MODEL;
        float a0 = 0.f, a1 = 0.f, a2 = 0.f, a3 = 0.f;
        if (s < 128) {
            const float* ir = inp + (((b*TSTEPS + tp)*129) + s + 1)*D_MODEL;
            for (int k = 0; k < D_MODEL; k += 4) {
                a0 += wr[k  ] * (pr[k  ] + ir[k  ]);
                a1 += wr[k+1] * (pr[k+1] + ir[k+1]);
                a2 += wr[k+2] * (pr[k+2] + ir[k+2]);
                a3 += wr[k+3] * (pr[k+3] + ir[k+3]);
            }
        } else {
            for (int k = 0; k < D_MODEL; k += 4) {
                a0 += wr[k  ] * pr[k  ];
                a1 += wr[k+1] * pr[k+1];
                a2 += wr[k+2] * pr[k+2];
                a3 += wr[k+3] * pr[k+3];
            }
        }
        float x = (a0 + a1) + (a2 + a3) + dt_bias[h];
        float dv = (x > 20.f) ? x : log1pf(expf(x));
        int gi = chunk*NPOS*NHEADS + item;
        dtv[gi]   = dv;
        logdA[gi] = dv * negA[h];
    }
}

// ---------------- K1b: per (b,h) suffix scan over 2048 positions -> w
__global__ void k1b_scan(float* __restrict__ ws) {
    const float* dtv   = ws + OFF_DTV;
    const float* logdA = ws + OFF_LOGDA;
    float* w = ws + OFF_W;
    int b = blockIdx.x / NHEADS;
    int h = blockIdx.x % NHEADS;
    int j = threadIdx.x;   // 0..31
    __shared__ float sm[32];
    const int L = TSTEPS*NPOS;
    long base = (long)b*L*NHEADS + h;
    float part = 0.f;
    for (int i = 0; i < 64; ++i) part += logdA[base + (long)(j*64 + i)*NHEADS];
    sm[j] = part;
    __syncthreads();
    float tail = 0.f;
    for (int jj = j + 1; jj < 32; ++jj) tail += sm[jj];
    float run = tail;
    for (int i = 63; i >= 0; --i) {
        long idx = base + (long)(j*64 + i)*NHEADS;
        w[idx] = dtv[idx] * expf(run);
        run += logdA[idx];
    }
}

// ---------------- K2: per-chunk projection GEMM + conv/SiLU + state GEMM
__global__ void __launch_bounds__(256) k2_chunk(
        const float* __restrict__ inp,
        const float* __restrict__ conv_w,
        const float* __restrict__ conv_b,
        float* __restrict__ ws) {
    extern __shared__ float lds[];
    float* stage = lds;                  // 256*64
    float* xw    = stage + NPOS*64;      // 32*256  (transposed: [p][s])
    float* wv    = xw + 32*NPOS;         // 256*24
    float* xwh   = wv + NPOS*NHEADS;     // 3*32
    float* bh    = xwh + 96;             // 3*64
    float* padc  = bh + 192;             // 64

    const float* pe      = ws + OFF_PE;
    const float* Wp      = ws + OFF_WP;
    const float* padproj = ws + OFF_PADPROJ;
    const float* w_all   = ws + OFF_W;
    float* Gint  = ws + OFF_GINT;
    float* Ghead = ws + OFF_GHEAD;

    int chunk = blockIdx.x;
    int b = chunk >> 3, tp = chunk & 7;
    int tid  = threadIdx.x;
    int wid  = tid >> 5;
    int lane = tid & 31;
    int lhalf = lane >> 4;
    int l16   = lane & 15;

    const float* perow  = pe + tp*D_MODEL;
    const float* inbase = inp + ((b*TSTEPS + tp)*129 + 1)*D_MODEL;

    // prefetch this chunk's input rows (128 rows * 192 f32 = 768 cachelines)
    for (int i = tid; i < 768; i += 256)
        __builtin_prefetch((const char*)inbase + (size_t)i*128, 0, 2);

    for (int i = tid; i < NPOS*NHEADS; i += 256) wv[i] = w_all[chunk*NPOS*NHEADS + i];
    for (int i = tid; i < NPOS*32; i += 256) xw[i] = 0.f;
    if (tid < 96) xwh[tid] = 0.f;

    for (int cb = 0; cb < 14; ++cb) {
        int base = (cb < 12) ? cb*64 : (768 + (cb - 12)*64);
        __syncthreads();
        if (tid < 64) padc[tid] = (tp > 0) ? padproj[(tp - 1)*PACKC + base + tid] : 0.f;

        // ---- projection GEMM over real rows only: stage[s<128][n] = frame @ Wp^T
        // rows 128..255 are the constant pad frame -> filled from padproj[tp].
        {
            int Mb = wid * 16;           // 8 waves x 16 rows = 128 rows
            v8f acc[4];
            for (int n = 0; n < 4; ++n) { v8f z = {}; acc[n] = z; }
            const float* arow = inbase + (Mb + l16)*D_MODEL;
            for (int k0 = 0; k0 < D_MODEL; k0 += 4) {
                int kk = k0 + 2*lhalf;
                v2f a;
                a[0] = perow[kk]     + arow[kk];
                a[1] = perow[kk + 1] + arow[kk + 1];
                for (int nt = 0; nt < 4; ++nt) {
                    const float* wr = Wp + (base + nt*16 + l16)*D_MODEL + kk;
                    v2f bb; bb[0] = wr[0]; bb[1] = wr[1];
                    acc[nt] = __builtin_amdgcn_wmma_f32_16x16x4_f32(
                        false, a, false, bb, (short)0, acc[nt], false, false);
                }
            }
            for (int nt = 0; nt < 4; ++nt)
                for (int r = 0; r < 8; ++r) {
                    int s = Mb + r + (lhalf << 3);
                    stage[s*64 + nt*16 + l16] = acc[nt][r];
                }
            // pad rows: constant per column
            for (int i = tid; i < 128*64; i += 256) {
                int s = 128 + (i >> 6), c = i & 63;
                stage[s*64 + c] = padproj[tp*PACKC + base + c];
            }
        }
        __syncthreads();

        if (cb < 12) {
            // x channels: conv + SiLU, accumulate head-weighted xw (+ head variant)
            int hbase = base >> 5;
            for (int pp = tid; pp < NPOS*32; pp += 256) {
                int s = pp >> 5, p = pp & 31;
                float accw = 0.f, accwh = 0.f;
                for (int hl = 0; hl < 2; ++hl) {
                    int cl = hl*32 + p;
                    int cg = base + cl;
                    float cw0 = conv_w[cg*4 + 0], cw1 = conv_w[cg*4 + 1];
                    float cw2 = conv_w[cg*4 + 2], cw3 = conv_w[cg*4 + 3];
                    float ctx = padc[cl];
                    float r0 = (s >= 3) ? stage[(s - 3)*64 + cl] : ctx;
                    float r1 = (s >= 2) ? stage[(s - 2)*64 + cl] : ctx;
                    float r2 = (s >= 1) ? stage[(s - 1)*64 + cl] : ctx;
                    float r3 = stage[s*64 + cl];
                    float cv = conv_b[cg] + r0*cw0 + r1*cw1 + r2*cw2 + r3*cw3;
                    float wgt = wv[s*NHEADS + hbase + hl];
                    accw += wgt * siluf(cv);
                    if (s < 3) {
                        float h0 = (s >= 3) ? r0 : 0.f;
                        float h1 = (s >= 2) ? r1 : 0.f;
                        float h2 = (s >= 1) ? r2 : 0.f;
                        float cvh = conv_b[cg] + h0*cw0 + h1*cw1 + h2*cw2 + r3*cw3;
                        accwh += wgt * siluf(cvh);
                    }
                }
                xw[p*NPOS + s] += accw;
                if (s < 3) xwh[s*32 + p] += accwh;
            }
        } else {
            // B-state half: conv + SiLU into registers, then overwrite stage
            int nb = cb - 12;
            float tmp[64];
            float headv = 0.f; int headidx = -1;
            for (int i = 0; i < 64; ++i) {
                int pp = i*256 + tid;
                int s = pp >> 6, nl = pp & 63;
                int cg = base + nl;
                float cw0 = conv_w[cg*4 + 0], cw1 = conv_w[cg*4 + 1];
                float cw2 = conv_w[cg*4 + 2], cw3 = conv_w[cg*4 + 3];
                float ctx = padc[nl];
                float r0 = (s >= 3) ? stage[(s - 3)*64 + nl] : ctx;
                float r1 = (s >= 2) ? stage[(s - 2)*64 + nl] : ctx;
                float r2 = (s >= 1) ? stage[(s - 1)*64 + nl] : ctx;
                float r3 = stage[s*64 + nl];
                float cv = conv_b[cg] + r0*cw0 + r1*cw1 + r2*cw2 + r3*cw3;
                tmp[i] = siluf(cv);
                if (s < 3) {
                    float h0 = (s >= 3) ? r0 : 0.f;
                    float h1 = (s >= 2) ? r1 : 0.f;
                    float h2 = (s >= 1) ? r2 : 0.f;
                    float cvh = conv_b[cg] + h0*cw0 + h1*cw1 + h2*cw2 + r3*cw3;
                    headv = siluf(cvh);
                    headidx = s*64 + nl;
                }
            }
            __syncthreads();
            for (int i = 0; i < 64; ++i) stage[i*256 + tid] = tmp[i];
            if (headidx >= 0) bh[headidx] = headv;
            __syncthreads();

            // ---- state GEMM: G[p][n] = sum_s xw[p][s] * B[s][n]  (M=32,N=64,K=256)
            {
                int mt = wid >> 2, nt = wid & 3;
                v8f acc = {};
                int p0 = mt*16 + l16;
                int n0 = nt*16 + l16;
                const float* ar = xw + p0*NPOS;
                for (int k0 = 0; k0 < NPOS; k0 += 4) {
                    int kk = k0 + 2*lhalf;
                    v2f a;  a[0]  = ar[kk];            a[1]  = ar[kk + 1];
                    v2f bb; bb[0] = stage[kk*64 + n0]; bb[1] = stage[(kk + 1)*64 + n0];
                    acc = __builtin_amdgcn_wmma_f32_16x16x4_f32(
                        false, a, false, bb, (short)0, acc, false, false);
                }
                for (int r = 0; r < 8; ++r) {
                    int p  = mt*16 + r + (lhalf << 3);
                    int nl = nt*16 + l16;
                    int ng = nb*64 + nl;
                    float g = acc[r];
                    float delta = 0.f;
                    for (int s = 0; s < 3; ++s)
                        delta += xwh[s*32 + p]*bh[s*64 + nl] - xw[p*NPOS + s]*stage[s*64 + nl];
                    long go = (long)chunk*HEADDIM*D_STATE + (long)p*D_STATE + ng;
                    Gint[go]  = g;
                    Ghead[go] = g + delta;
                }
            }
        }
    }
}

// ---------------- K4: suffix-sum over chunks + classifier
__global__ void __launch_bounds__(256) k4_combine(
        const float* __restrict__ cls_w,
        const float* __restrict__ cls_b,
        const float* __restrict__ ws,
        float* __restrict__ out) {
    extern __shared__ float sm[];
    float* aggt    = sm;          // 4096
    float* running = sm + 4096;   // 4096
    __shared__ float red[256];
    const float* Gint  = ws + OFF_GINT;
    const float* Ghead = ws + OFF_GHEAD;
    int b = blockIdx.x, tid = threadIdx.x;
    int j = tid & 127, seg = tid >> 7;   // 2 K-segments of 2048
    for (int f = tid; f < 4096; f += 256) running[f] = 0.f;
    __syncthreads();
    for (int t = TSTEPS - 1; t >= 0; --t) {
        long gb = (long)(b*TSTEPS + t)*4096;
        for (int f = tid; f < 4096; f += 256) aggt[f] = Ghead[gb + f] + running[f];
        __syncthreads();
        float a0 = 0.f, a1 = 0.f, a2 = 0.f, a3 = 0.f;
        if (j < 100) {
            const float* wr = cls_w + (size_t)j*4096 + seg*2048;
            const float* ar = aggt + seg*2048;
            for (int f = 0; f < 2048; f += 4) {
                a0 += ar[f  ] * wr[f  ];
                a1 += ar[f+1] * wr[f+1];
                a2 += ar[f+2] * wr[f+2];
                a3 += ar[f+3] * wr[f+3];
            }
        }
        red[tid] = (a0 + a1) + (a2 + a3);
        __syncthreads();
        if (tid < 100) out[(b*TSTEPS + t)*100 + tid] = cls_b[tid] + red[tid] + red[tid + 128];
        for (int f = tid; f < 4096; f += 256) running[f] += Gint[gb + f];
        __syncthreads();
    }
}

extern "C" void kernel_launch(void* const* d_in, const int* in_sizes, int n_in,
                              void* d_out, int out_size, void* d_ws, size_t ws_size,
                              hipStream_t stream) {
    (void)in_sizes; (void)n_in; (void)out_size; (void)ws_size;
    const float* inputs    = (const float*)d_in[0];
    const float* in_proj_w = (const float*)d_in[1];
    const float* conv_w    = (const float*)d_in[2];
    const float* conv_b    = (const float*)d_in[3];
    const float* dt_bias   = (const float*)d_in[4];
    const float* A_log     = (const float*)d_in[5];
    const float* cls_w     = (const float*)d_in[6];
    const float* cls_b     = (const float*)d_in[7];
    float* out = (float*)d_out;
    float* ws  = (float*)d_ws;

    k0_setup<<<1, 256, 0, stream>>>(in_proj_w, A_log, ws);
    k1_dt<<<NCHUNK, 256, 0, stream>>>(inputs, dt_bias, ws);
    k1b_scan<<<BATCH*NHEADS, 32, 0, stream>>>(ws);
    size_t lds2 = (size_t)(NPOS*64 + 32*NPOS + NPOS*NHEADS + 96 + 192 + 64) * sizeof(float);
    k2_chunk<<<NCHUNK, 256, lds2, stream>>>(inputs, conv_w, conv_b, ws);
    k4_combine<<<BATCH, 256, 8192*sizeof(float), stream>>>(cls_w, cls_b, ws, out);
}